// BoundaryLoss_39943195853299
// MI455X (gfx1250) — compile-verified
//
#include <hip/hip_runtime.h>
#include <hip/hip_bf16.h>
#include <math.h>

// BoundaryLoss: exact 2D EDT (column scan + O(W^2) row min-plus) -> boundary
// weights -> weighted BCE-with-logits -> global mean. B=8, H=W=384.
// Deterministic fixed-order reductions (no atomics). Wave reductions use
// V_WMMA_F32_16X16X4_F32 with an all-ones B matrix (exact f32 FMA math).

#define Bimg 8
#define Hdim 384
#define Wdim 384
#define Ntot (Bimg * Hdim * Wdim)      // 1179648
#define NROWS (Bimg * Hdim)            // 3072
#define BIGF 1.0e9f
// LDS row stride: 388 dwords == 4 (mod 64 banks) so the two class rows never
// collide in a bank; 388*4 bytes is 16B-aligned so float4/ds_load_b128 works.
#define SSTRIDE 388

typedef __attribute__((ext_vector_type(2))) float v2f;
typedef __attribute__((ext_vector_type(8))) float v8f;

// Exact full-wave (32-lane) sum via one WMMA:
//   A (16x4): VGPR0 = p (lane m -> row m%16), VGPR1 = 0
//   B (4x16): all ones
//   D(m,n) = sum_k A(m,k) = p[m] + p[m+16]  (same for every column n)
// Lanes 0-15 hold rows 0..7 in the 8 accumulators, lanes 16-31 rows 8..15, so
// summing the 8 D VGPRs per lane and adding lane0 + lane16 gives the wave sum.
// Requires EXEC all-ones: call only from uniform control flow in full blocks.
__device__ __forceinline__ float wave_sum_wmma(float p) {
    v2f a; a.x = p;    a.y = 0.0f;
    v2f b; b.x = 1.0f; b.y = 1.0f;
    v8f c = {};
    c = __builtin_amdgcn_wmma_f32_16x16x4_f32(
        /*neg_a=*/false, a, /*neg_b=*/false, b,
        /*c_mod=*/(short)0, c, /*reuse_a=*/false, /*reuse_b=*/false);
    float e = ((c[0] + c[1]) + (c[2] + c[3])) + ((c[4] + c[5]) + (c[6] + c[7]));
    return __shfl(e, 0, 32) + __shfl(e, 16, 32);
}

// Kernel 1: per-column forward/backward min-scan (matches the jax scan in f32
// exactly: min(c+1, x) with BIG=1e9 stays exactly 1e9 in f32). Writes the
// SQUARED vertical distance for both masks.
__global__ __launch_bounds__(256) void col_scan_kernel(
    const float* __restrict__ targ,
    float* __restrict__ gsq_pos,   // mask: t <= 0.5
    float* __restrict__ gsq_neg)   // mask: t >  0.5
{
    int tid = blockIdx.x * blockDim.x + threadIdx.x;
    if (tid >= Bimg * Wdim) return;
    int b = tid / Wdim, j = tid % Wdim;
    int base = b * Hdim * Wdim + j;

    float cp = BIGF, cn = BIGF;
    for (int i = 0; i < Hdim; ++i) {
        int idx = base + i * Wdim;
        float tv = targ[idx];
        float xp = (tv <= 0.5f) ? 0.0f : BIGF;
        float xn = (tv >  0.5f) ? 0.0f : BIGF;
        cp = fminf(cp + 1.0f, xp);
        cn = fminf(cn + 1.0f, xn);
        gsq_pos[idx] = cp;          // temporarily holds forward distance
        gsq_neg[idx] = cn;
    }
    cp = BIGF; cn = BIGF;
    for (int i = Hdim - 1; i >= 0; --i) {
        int idx = base + i * Wdim;
        float tv = targ[idx];
        float xp = (tv <= 0.5f) ? 0.0f : BIGF;
        float xn = (tv >  0.5f) ? 0.0f : BIGF;
        cp = fminf(cp + 1.0f, xp);
        cn = fminf(cn + 1.0f, xn);
        float gp = fminf(cp, gsq_pos[idx]);
        float gn = fminf(cn, gsq_neg[idx]);
        gsq_pos[idx] = gp * gp;     // final: squared vertical distance
        gsq_neg[idx] = gn * gn;
    }
}

// Kernel 2: one block per (image,row). Stage both squared-distance rows in LDS,
// each pixel does the 384-candidate min-plus against the opposite class's row
// (float4 / ds_load_b128, bank-conflict-free), then weight * BCE, then exact
// WMMA tree reduction to one partial per row.
__global__ __launch_bounds__(Wdim) void row_edt_bce_kernel(
    const float* __restrict__ pred,
    const float* __restrict__ targ,
    const float* __restrict__ gsq_pos,
    const float* __restrict__ gsq_neg,
    float* __restrict__ partials)
{
    __shared__ __align__(16) float srow[2 * SSTRIDE];  // [0..]=pos, [SSTRIDE..]=neg
    __shared__ float wsum[Wdim / 32];                  // 12 wave sums

    const int row  = blockIdx.x;             // 0..NROWS-1  (= b*H + i)
    const int j    = threadIdx.x;            // 0..383
    const int base = row * Wdim;

    srow[j]           = gsq_pos[base + j];
    srow[SSTRIDE + j] = gsq_neg[base + j];
    float t = targ[base + j];
    float p = pred[base + j];
    if (row + 1 < NROWS) {                   // warm L2/WGP$ for the next row
        __builtin_prefetch(&gsq_pos[base + Wdim + j], 0, 0);
        __builtin_prefetch(&gsq_neg[base + Wdim + j], 0, 0);
    }
    __syncthreads();

    // |dist| at this pixel uses the opposite-class EDT:
    //   t>0.5 -> EDT of (t<=0.5) = srow[0..];  else srow[SSTRIDE..]
    const float* s = (t > 0.5f) ? &srow[0] : &srow[SSTRIDE];
    const float fj = (float)j;
    float d2a = 3.402823466e38f, d2b = 3.402823466e38f;
    float f0 = fj;                           // == fj - k, exact small ints
    #pragma unroll 4
    for (int k = 0; k < Wdim; k += 4) {
        float4 g = *reinterpret_cast<const float4*>(s + k);
        float f1 = f0 - 1.0f;
        float f2 = f0 - 2.0f;
        float f3 = f0 - 3.0f;
        d2a = fminf(d2a, fmaf(f0, f0, g.x));
        d2b = fminf(d2b, fmaf(f1, f1, g.y));
        d2a = fminf(d2a, fmaf(f2, f2, g.z));
        d2b = fminf(d2b, fmaf(f3, f3, g.w));
        f0 -= 4.0f;
    }
    float d = sqrtf(fminf(d2a, d2b));
    // w = sigmoid(-(d - 3)/5) = 1/(1 + exp((d-3)*0.2)); exp overflow -> w=0, ok
    float w = 1.0f / (1.0f + expf((d - 3.0f) * 0.2f));

    // Stable BCE-with-logits: log_sigmoid(x) = min(x,0) - log1p(exp(-|x|))
    float ap  = fabsf(p);
    float l1p = log1pf(expf(-ap));
    float lsp = fminf(p, 0.0f) - l1p;        // log_sigmoid(p)
    float lsn = fminf(-p, 0.0f) - l1p;       // log_sigmoid(-p)
    float bce = -(t * lsp + (1.0f - t) * lsn);
    float val = bce * w;

    // Stage 1: exact per-wave sum via WMMA (all 12 waves full, EXEC all-ones)
    float wsv = wave_sum_wmma(val);
    if ((j & 31) == 0) wsum[j >> 5] = wsv;
    __syncthreads();

    // Stage 2: every wave redundantly reduces the 12 wave sums (keeps EXEC
    // all-ones for the WMMA); thread 0 writes the row partial.
    int lane = j & 31;
    float v = (lane < (Wdim / 32)) ? wsum[lane] : 0.0f;
    float tot = wave_sum_wmma(v);
    if (j == 0) partials[row] = tot;
}

// Kernel 3: single block sums the 3072 row partials (fixed order) and writes
// the mean.
__global__ __launch_bounds__(256) void final_reduce_kernel(
    const float* __restrict__ partials,
    float* __restrict__ out)
{
    __shared__ float wsum[8];
    const int tid = threadIdx.x;

    float s = 0.0f;
    for (int i = tid; i < NROWS; i += 256) s += partials[i];

    float wsv = wave_sum_wmma(s);
    if ((tid & 31) == 0) wsum[tid >> 5] = wsv;
    __syncthreads();

    int lane = tid & 31;
    float v = (lane < 8) ? wsum[lane] : 0.0f;
    float tot = wave_sum_wmma(v);
    if (tid == 0) out[0] = tot * (1.0f / (float)Ntot);
}

extern "C" void kernel_launch(void* const* d_in, const int* in_sizes, int n_in,
                              void* d_out, int out_size, void* d_ws, size_t ws_size,
                              hipStream_t stream) {
    const float* pred = (const float*)d_in[0];   // predictions (8,1,384,384) f32
    const float* targ = (const float*)d_in[1];   // targets     (8,1,384,384) f32

    float* ws       = (float*)d_ws;
    float* gsq_pos  = ws;                        // N floats
    float* gsq_neg  = ws + Ntot;                 // N floats
    float* partials = ws + 2 * Ntot;             // NROWS floats

    col_scan_kernel<<<(Bimg * Wdim + 255) / 256, 256, 0, stream>>>(
        targ, gsq_pos, gsq_neg);
    row_edt_bce_kernel<<<NROWS, Wdim, 0, stream>>>(
        pred, targ, gsq_pos, gsq_neg, partials);
    final_reduce_kernel<<<1, 256, 0, stream>>>(partials, (float*)d_out);
}